// GraphPooling_76888504533410
// MI455X (gfx1250) — compile-verified
//
#include <hip/hip_runtime.h>
#include <hip/hip_bf16.h>

typedef __attribute__((ext_vector_type(2))) float v2f;
typedef __attribute__((ext_vector_type(8))) float v8f;

#define NF_E 256
#define HIDDEN 128
#define NH 4
#define NODES_PER_GRAPH 256
#define K_DIM 256
#define BLK_NODES 64          // nodes per block (4 groups of 16)
#define NGROUPS (BLK_NODES / 16)

// ---------------------------------------------------------------------------
// Kernel 1: per-graph query projection folded with b1:
//   q1[b,h,k] = b1[h,k] + sum_{j<256} nf[head_b,j]*W1[h,256+j,k]
//                       + sum_{j<256} nf[tail_b,j]*W1[h,512+j,k]
// grid (B, H), block 128 (one thread per k)
// ---------------------------------------------------------------------------
__global__ void __launch_bounds__(128) qproj_kernel(
    const float* __restrict__ nf, const float* __restrict__ W1,
    const float* __restrict__ b1, const int* __restrict__ head_ids,
    const int* __restrict__ tail_ids, float* __restrict__ q1) {
  const int b = blockIdx.x;
  const int h = blockIdx.y;
  const int k = threadIdx.x;  // 0..127
  const float* w = W1 + h * (768 * HIDDEN) + 256 * HIDDEN + k;
  const float* hrow = nf + (size_t)head_ids[b] * NF_E;
  const float* trow = nf + (size_t)tail_ids[b] * NF_E;
  float acc = b1[h * HIDDEN + k];
  #pragma unroll 4
  for (int j = 0; j < 256; ++j) acc = fmaf(hrow[j], w[j * HIDDEN], acc);
  w += 256 * HIDDEN;
  #pragma unroll 4
  for (int j = 0; j < 256; ++j) acc = fmaf(trow[j], w[j * HIDDEN], acc);
  q1[(b * NH + h) * HIDDEN + k] = acc;
}

// ---------------------------------------------------------------------------
// Kernel 2: per-node MLP scores via WMMA f32 16x16x4.
//   scores[n,h] = b2[h] + sum_k relu( nf[n,:]@W1[h,0:256,k] + q1[g(n),h,k] ) * W2[h,k]
// Block = 256 threads (8 wave32s), owns 64 nodes (all within one graph).
// Wave w owns columns [w*64, w*64+64) of the 512-wide (H*HID) output:
//   head h = w>>1, within-head col base kpBase = (w&1)*64, 4 tiles of 16 cols.
// A-tile (64x256 f32 = 64KB) staged via GLOBAL_LOAD_ASYNC_TO_LDS_B128 with an
// XOR swizzle (c4 ^ (row&15)) -> bank-conflict-free fragment reads at stride
// 256 (no padding needed, LDS exactly 64KB).
// ---------------------------------------------------------------------------
__global__ void __launch_bounds__(256) scores_kernel(
    const float* __restrict__ nf, const float* __restrict__ W1,
    const float* __restrict__ W2, const float* __restrict__ b2,
    const float* __restrict__ q1, float* __restrict__ scores) {
  __shared__ float ldsA[BLK_NODES * NF_E];   // 64 KB, XOR-swizzled
  __shared__ float ldsP[BLK_NODES * NH];     // 1 KB partial scores

  const int tid = threadIdx.x;
  const int nodeBase = blockIdx.x * BLK_NODES;
  const int graph = nodeBase >> 8;  // 256 nodes per graph, blocks never straddle

  // Async-stage 64x256 node-feature tile into LDS (16B chunks, swizzled).
  #pragma unroll
  for (int i = tid; i < BLK_NODES * (NF_E / 4); i += 256) {
    const int row = i >> 6;        // 64 float4 per row
    const int c4 = i & 63;
    const int sw = c4 ^ (row & 15);
    const unsigned long long ga =
        (unsigned long long)(uintptr_t)(nf + (size_t)(nodeBase + row) * NF_E +
                                        c4 * 4);
    const unsigned int la =
        (unsigned int)(uintptr_t)(&ldsA[row * NF_E + sw * 4]);
    asm volatile("global_load_async_to_lds_b128 %0, %1, off"
                 :
                 : "v"(la), "v"(ga)
                 : "memory");
  }
  if (tid < BLK_NODES * NH) ldsP[tid] = 0.0f;
  asm volatile("s_wait_asynccnt 0x0" ::: "memory");
  __syncthreads();

  const int wave = tid >> 5;
  const int lane = tid & 31;
  const int m = lane & 15;              // row within 16 (A) / col within 16 (B)
  const int koff = (lane >> 4) * 2;     // K sub-offset: 0 (lanes 0-15) / 2 (16-31)
  const int h = wave >> 1;              // head for this wave
  const int kpBase = (wave & 1) * 64;   // within-head column base

  v8f acc[NGROUPS][4] = {};  // [node-group][col-tile]

  // B source pointers: column = h*128 + kpBase + t*16 + m, rows 0..255 of W1.
  const float* bptr[4];
  #pragma unroll
  for (int t = 0; t < 4; ++t)
    bptr[t] = W1 + h * (768 * HIDDEN) + kpBase + t * 16 + m;

  #pragma unroll 2
  for (int kk = 0; kk < K_DIM; kk += 4) {
    const int kI = kk + koff;
    const int swc = ((kk >> 2) ^ m) * 4 + koff;  // swizzled column (floats)
    // A fragments (16x4 layout): lane holds {K=kI, K=kI+1} of row g*16+m.
    v2f a[NGROUPS];
    #pragma unroll
    for (int g = 0; g < NGROUPS; ++g)
      a[g] = *(const v2f*)(&ldsA[(g * 16 + m) * NF_E + swc]);
    #pragma unroll
    for (int t = 0; t < 4; ++t) {
      v2f bf;  // B fragment (4x16): VGPR0 -> K=kI, VGPR1 -> K=kI+1, col m
      bf[0] = bptr[t][(kI)*HIDDEN];
      bf[1] = bptr[t][(kI + 1) * HIDDEN];
      #pragma unroll
      for (int g = 0; g < NGROUPS; ++g)
        acc[g][t] = __builtin_amdgcn_wmma_f32_16x16x4_f32(
            false, a[g], false, bf, (short)0, acc[g][t], false, false);
    }
  }

  // Epilogue: bias(q1) + ReLU + dot with W2, reduced across the 16 output cols
  // of each tile via cross-lane shuffles, accumulated across waves in LDS.
  #pragma unroll
  for (int t = 0; t < 4; ++t) {
    const int col = kpBase + t * 16 + m;  // within-head column this lane owns
    const float q1v = q1[(graph * NH + h) * HIDDEN + col];
    const float w2v = W2[h * HIDDEN + col];
    #pragma unroll
    for (int g = 0; g < NGROUPS; ++g) {
      #pragma unroll
      for (int r = 0; r < 8; ++r) {
        float v = fmaxf(acc[g][t][r] + q1v, 0.0f) * w2v;
        v += __shfl_xor(v, 8, 16);
        v += __shfl_xor(v, 4, 16);
        v += __shfl_xor(v, 2, 16);
        v += __shfl_xor(v, 1, 16);
        if (m == 0) {  // lane 0 -> M=r, lane 16 -> M=r+8
          const int nodeLocal = g * 16 + r + ((lane >> 4) << 3);
          atomicAdd(&ldsP[nodeLocal * NH + h], v);
        }
      }
    }
  }
  __syncthreads();
  {
    const int nodeLocal = tid >> 2;  // 0..63
    const int hh = tid & 3;
    scores[(size_t)(nodeBase + nodeLocal) * NH + hh] = ldsP[tid] + b2[hh];
  }
}

// ---------------------------------------------------------------------------
// Kernel 3: per-graph segment softmax over heads + weighted pooling.
// out[b,e] = sum_n ( (1/H) * sum_h alpha[n,h] ) * nf[n,e]
// One block per graph, 256 threads.
// ---------------------------------------------------------------------------
__global__ void __launch_bounds__(256) pool_kernel(
    const float* __restrict__ nf, const float* __restrict__ scores,
    float* __restrict__ out) {
  __shared__ float4 red[256];
  __shared__ float wn[256];
  const int t = threadIdx.x;
  const int base = blockIdx.x * NODES_PER_GRAPH;

  const float4 s = *(const float4*)(scores + (size_t)(base + t) * NH);

  // segment max per head
  red[t] = s;
  __syncthreads();
  for (int off = 128; off > 0; off >>= 1) {
    if (t < off) {
      float4 c = red[t];
      const float4 o = red[t + off];
      c.x = fmaxf(c.x, o.x); c.y = fmaxf(c.y, o.y);
      c.z = fmaxf(c.z, o.z); c.w = fmaxf(c.w, o.w);
      red[t] = c;
    }
    __syncthreads();
  }
  const float4 mx = red[0];
  __syncthreads();

  float4 ex;
  ex.x = expf(s.x - mx.x);
  ex.y = expf(s.y - mx.y);
  ex.z = expf(s.z - mx.z);
  ex.w = expf(s.w - mx.w);

  // segment sum per head
  red[t] = ex;
  __syncthreads();
  for (int off = 128; off > 0; off >>= 1) {
    if (t < off) {
      float4 c = red[t];
      const float4 o = red[t + off];
      c.x += o.x; c.y += o.y; c.z += o.z; c.w += o.w;
      red[t] = c;
    }
    __syncthreads();
  }
  const float4 sm = red[0];

  wn[t] = 0.25f * (ex.x / sm.x + ex.y / sm.y + ex.z / sm.z + ex.w / sm.w);
  __syncthreads();

  // weighted column sum: thread t owns output feature e = t
  const float* nfb = nf + (size_t)base * NF_E;
  float acc = 0.0f;
  #pragma unroll 4
  for (int n = 0; n < NODES_PER_GRAPH; ++n)
    acc = fmaf(wn[n], nfb[(size_t)n * NF_E + t], acc);
  out[(size_t)blockIdx.x * NF_E + t] = acc;
}

// ---------------------------------------------------------------------------
// Launch: inputs are (node_features, W1, b1, W2, b2, graph_ids, head_ids,
// tail_ids); output [B=512, E=256] float32.
// ---------------------------------------------------------------------------
extern "C" void kernel_launch(void* const* d_in, const int* in_sizes, int n_in,
                              void* d_out, int out_size, void* d_ws,
                              size_t ws_size, hipStream_t stream) {
  (void)in_sizes; (void)n_in; (void)out_size; (void)ws_size;
  const float* nf = (const float*)d_in[0];
  const float* W1 = (const float*)d_in[1];
  const float* b1 = (const float*)d_in[2];
  const float* W2 = (const float*)d_in[3];
  const float* b2 = (const float*)d_in[4];
  const int* head_ids = (const int*)d_in[6];
  const int* tail_ids = (const int*)d_in[7];
  float* out = (float*)d_out;

  const int N = 131072, B = 512;
  float* q1 = (float*)d_ws;                      // [B, H, HID] = 1 MB
  float* scores = q1 + (size_t)B * NH * HIDDEN;  // [N, H]      = 2 MB

  qproj_kernel<<<dim3(B, NH), 128, 0, stream>>>(nf, W1, b1, head_ids, tail_ids,
                                                q1);
  scores_kernel<<<dim3(N / BLK_NODES), 256, 0, stream>>>(nf, W1, W2, b2, q1,
                                                         scores);
  pool_kernel<<<dim3(B), 256, 0, stream>>>(nf, scores, out);
}